// SpaMamba_19112604467644
// MI455X (gfx1250) — compile-verified
//
#include <hip/hip_runtime.h>
#include <hip/hip_bf16.h>
#include <math.h>

// ---------------- problem constants ----------------
#define BB   2
#define CC   64
#define HH   128
#define WW   128
#define LL   (HH*WW)        // 16384
#define DI   128            // D_INNER
#define EE   (2*DI)         // 256
#define DS   16             // D_STATE
#define DTR  4              // DT_RANK
#define NXP  48             // x_proj output padded 36 -> 48 (3 WMMA n-tiles)
#define GEPS 1e-5f

typedef float v2f __attribute__((ext_vector_type(2)));
typedef float v8f __attribute__((ext_vector_type(8)));

// D = A(16x4 f32) * B(4x16 f32) + C(16x16 f32)   -- full fp32 WMMA path
__device__ __forceinline__ v8f wmma_k4(v2f a, v2f b, v8f c) {
    return __builtin_amdgcn_wmma_f32_16x16x4_f32(
        /*neg_a=*/false, a, /*neg_b=*/false, b,
        /*c_mod=*/(short)0, c, /*reuse_a=*/false, /*reuse_b=*/false);
}

__device__ __forceinline__ float silu_f(float v) {
    return v / (1.f + __expf(-v));
}

// ---- CDNA5 async global->LDS copies (ASYNCcnt path, no VGPR round-trip) ----
// LDS offset: low 32 bits of the flat shared-pointer address (ISA 10.2)
// the offset within the workgroup LDS allocation.
__device__ __forceinline__ unsigned lds_off(const void* p) {
    return (unsigned)(uintptr_t)p;
}
__device__ __forceinline__ void async_g2l_b128(unsigned loff, const void* g) {
    asm volatile("global_load_async_to_lds_b128 %0, %1, off"
                 :: "v"(loff), "v"((unsigned long long)(uintptr_t)g) : "memory");
}
__device__ __forceinline__ void async_g2l_b32(unsigned loff, const void* g) {
    asm volatile("global_load_async_to_lds_b32 %0, %1, off"
                 :: "v"(loff), "v"((unsigned long long)(uintptr_t)g) : "memory");
}
__device__ __forceinline__ void wait_async0() {
    asm volatile("s_wait_asynccnt 0x0" ::: "memory");
}

// ---------------- workspace layout (floats) ----------------
#define OFF_XZ   ((size_t)0)                       // (B,L,256)   8388608
#define OFF_U    (OFF_XZ  + (size_t)BB*LL*EE)      // (B,L,128)   4194304
#define OFF_DBC  (OFF_U   + (size_t)BB*LL*DI)      // (B,L,48)    1572864
#define OFF_DT   (OFF_DBC + (size_t)BB*LL*NXP)     // (B,L,128)   4194304
#define OFF_YG   (OFF_DT  + (size_t)BB*LL*DI)      // (B,L,128)   4194304
#define OFF_YO   (OFF_YG  + (size_t)BB*LL*DI)      // (B,L,64)    2097152
#define OFF_GN   (OFF_YO  + (size_t)BB*LL*CC)      // 16 floats (mu,rstd)*8

// =====================================================================
// Kernel 1: in_proj  XZ[b,l,e] = sum_c X[b,c,l] * W[e,c]
//  W (256x64, 64KB) and both 16x64 X tiles async-staged into LDS.
//  wave = 16 L-rows x 64 E-cols.
// =====================================================================
__global__ void k_inproj(const float* __restrict__ X,
                         const float* __restrict__ W,
                         float* __restrict__ XZ) {
    __shared__ float Wl[EE * CC];                 // 64 KB
    __shared__ float XT[2][CC * 16];              // 2 tiles x (c-major, 16 l) = 8 KB

    // async-stage weights (b128 per lane)
    const unsigned wbase = lds_off(&Wl[0]);
    for (int i = threadIdx.x * 4; i < EE * CC; i += blockDim.x * 4)
        async_g2l_b128(wbase + i * 4, W + i);

    // async-stage the two X tiles of this block (column-coalesced b32)
    const unsigned xbase = lds_off(&XT[0][0]);
    for (int i = threadIdx.x; i < 2 * CC * 16; i += blockDim.x) {
        const int tl = i >> 10;                   // tile within block (CC*16==1024)
        const int c  = (i >> 4) & (CC - 1);
        const int r  = i & 15;
        const int tile = blockIdx.x * 2 + tl;
        const int bi = tile >> 10;
        const int l0 = (tile & 1023) << 4;
        async_g2l_b32(xbase + i * 4, X + ((size_t)bi * CC + c) * LL + l0 + r);
    }
    wait_async0();
    __syncthreads();

    const int wave = threadIdx.x >> 5;
    const int lane = threadIdx.x & 31;
    const int half = lane >> 4, lm = lane & 15;
    const int tile = blockIdx.x * 2 + (wave >> 2);   // 2048 tiles total
    const int e0   = (wave & 3) * 64;
    const int bi   = tile >> 10;
    const int l0   = (tile & 1023) << 4;
    const float* Xt = &XT[wave >> 2][0];

    v8f acc[4];
#pragma unroll
    for (int n = 0; n < 4; ++n) acc[n] = (v8f)0.f;

    for (int kk = 0; kk < CC; kk += 4) {
        const int kc = kk + half * 2;
        v2f a;
        a.x = Xt[kc * 16 + lm];
        a.y = Xt[(kc + 1) * 16 + lm];
#pragma unroll
        for (int n = 0; n < 4; ++n) {
            const int e = e0 + n * 16 + lm;
            v2f bf;
            bf.x = Wl[e * CC + kc];
            bf.y = Wl[e * CC + kc + 1];
            acc[n] = wmma_k4(a, bf, acc[n]);
        }
    }
#pragma unroll
    for (int n = 0; n < 4; ++n)
#pragma unroll
        for (int r = 0; r < 8; ++r) {
            const int row = l0 + r + half * 8;
            const int col = e0 + n * 16 + lm;
            XZ[((size_t)bi * LL + row) * EE + col] = acc[n][r];
        }
}

// =====================================================================
// Kernel 2: depthwise causal conv (D_CONV=4) + SiLU -> U (B,L,128)
// =====================================================================
__global__ void k_conv(const float* __restrict__ XZ,
                       const float* __restrict__ cw,
                       const float* __restrict__ cb,
                       float* __restrict__ U) {
    const int idx = blockIdx.x * blockDim.x + threadIdx.x;
    if (idx >= BB * LL * DI) return;
    const int d  = idx & (DI - 1);
    const int l  = (idx >> 7) & (LL - 1);
    const int bi = idx >> 21;
    float s = cb[d];
#pragma unroll
    for (int k = 0; k < 4; ++k) {
        const int ls = l + k - 3;
        if (ls >= 0) s += cw[d * 4 + k] * XZ[((size_t)bi * LL + ls) * EE + d];
    }
    U[idx] = silu_f(s);
}

// =====================================================================
// Kernel 3: x_proj  DBC[b,l,0:36] = U(b,l,:) @ x_proj_w^T (padded to 48)
//  weights async-staged (zero-padded 48x128) in LDS; wave = 16 x 48.
// =====================================================================
__global__ void k_xproj(const float* __restrict__ U,
                        const float* __restrict__ XPW,
                        float* __restrict__ DBC) {
    __shared__ float Wl[NXP * DI];                // 24 KB
    const unsigned wbase = lds_off(&Wl[0]);
    for (int i = threadIdx.x * 4; i < (DTR + 2 * DS) * DI; i += blockDim.x * 4)
        async_g2l_b128(wbase + i * 4, XPW + i);
    for (int i = (DTR + 2 * DS) * DI + threadIdx.x; i < NXP * DI; i += blockDim.x)
        Wl[i] = 0.f;                              // zero pad rows 36..47
    wait_async0();
    __syncthreads();

    const int wave = threadIdx.x >> 5;
    const int lane = threadIdx.x & 31;
    const int half = lane >> 4, lm = lane & 15;
    const int tile = blockIdx.x * 8 + wave;
    const int bi = tile >> 10, l0 = (tile & 1023) << 4;

    v8f acc[3];
#pragma unroll
    for (int n = 0; n < 3; ++n) acc[n] = (v8f)0.f;

    const float* Ub = U + (size_t)bi * LL * DI;
    for (int kk = 0; kk < DI; kk += 4) {
        const int kc = kk + half * 2;
        v2f a;
        a.x = Ub[(size_t)(l0 + lm) * DI + kc];
        a.y = Ub[(size_t)(l0 + lm) * DI + kc + 1];
#pragma unroll
        for (int n = 0; n < 3; ++n) {
            v2f bf;
            bf.x = Wl[(n * 16 + lm) * DI + kc];
            bf.y = Wl[(n * 16 + lm) * DI + kc + 1];
            acc[n] = wmma_k4(a, bf, acc[n]);
        }
    }
#pragma unroll
    for (int n = 0; n < 3; ++n)
#pragma unroll
        for (int r = 0; r < 8; ++r)
            DBC[((size_t)bi * LL + l0 + r + half * 8) * NXP + n * 16 + lm] = acc[n][r];
}

// =====================================================================
// Kernel 4: dt = softplus(dt_r @ dt_proj_w^T + dt_proj_b)  (K=4 -> one WMMA)
// =====================================================================
__global__ void k_dtproj(const float* __restrict__ DBC,
                         const float* __restrict__ DTW,
                         const float* __restrict__ DTB,
                         float* __restrict__ DT) {
    const int wave = threadIdx.x >> 5;
    const int lane = threadIdx.x & 31;
    const int half = lane >> 4, lm = lane & 15;
    const int tile = blockIdx.x * 8 + wave;
    const int bi = tile >> 10, l0 = (tile & 1023) << 4;

    const int kc = half * 2;
    v2f a;
    a.x = DBC[((size_t)bi * LL + l0 + lm) * NXP + kc];
    a.y = DBC[((size_t)bi * LL + l0 + lm) * NXP + kc + 1];

    v8f acc[8];
#pragma unroll
    for (int n = 0; n < 8; ++n) {
        acc[n] = (v8f)0.f;
        const int e = n * 16 + lm;
        v2f bf;
        bf.x = DTW[e * DTR + kc];
        bf.y = DTW[e * DTR + kc + 1];
        acc[n] = wmma_k4(a, bf, acc[n]);
    }
#pragma unroll
    for (int n = 0; n < 8; ++n) {
        const float bias = DTB[n * 16 + lm];
#pragma unroll
        for (int r = 0; r < 8; ++r) {
            float v = acc[n][r] + bias;
            float sp = (v > 20.f) ? v : log1pf(__expf(v));   // softplus
            DT[((size_t)bi * LL + l0 + r + half * 8) * DI + n * 16 + lm] = sp;
        }
    }
}

// =====================================================================
// Kernel 5: selective scan + gating (latency-bound serial chain).
//  wave = 2 channels x 16 states; lane = (d_half, s). 128 waves total.
//  h_t = exp(dt*A)*h + dt*u*B_t ;  y_t = <h, C_t> ; yg = (y+u*D)*silu(z)
//  global_prefetch_b8 issued 64 steps ahead to hide HBM latency.
// =====================================================================
__global__ void k_scan(const float* __restrict__ DT,
                       const float* __restrict__ U,
                       const float* __restrict__ DBC,
                       const float* __restrict__ XZ,
                       const float* __restrict__ Alog,
                       const float* __restrict__ Dp,
                       float* __restrict__ YG) {
    const int gtid = blockIdx.x * blockDim.x + threadIdx.x;
    const int wv   = gtid >> 5;          // 0..127
    const int lane = gtid & 31;
    const int s = lane & 15, dh = lane >> 4;
    const int bi = wv >> 6;
    const int d  = ((wv & 63) << 1) + dh;

    const float Ads = -__expf(Alog[d * DS + s]);
    const float Dd  = Dp[d];
    float h = 0.f;

    const float* dtb  = DT  + (size_t)bi * LL * DI + d;
    const float* ub   = U   + (size_t)bi * LL * DI + d;
    const float* dbcb = DBC + (size_t)bi * LL * NXP;
    const float* zb   = XZ  + (size_t)bi * LL * EE + DI + d;
    float*       ygb  = YG  + (size_t)bi * LL * DI + d;

    const int PF = 64;                    // prefetch distance (steps)
#pragma unroll 4
    for (int t = 0; t < LL; ++t) {
        // speculative prefetch of the t+PF operand rows (safe past end: ws-mapped)
        __builtin_prefetch(dtb  + (size_t)(t + PF) * DI, 0, 1);
        __builtin_prefetch(ub   + (size_t)(t + PF) * DI, 0, 1);
        __builtin_prefetch(dbcb + (size_t)(t + PF) * NXP + s, 0, 1);

        const float dtv = dtb[(size_t)t * DI];
        const float uv  = ub[(size_t)t * DI];
        const float Bv  = dbcb[(size_t)t * NXP + DTR + s];
        const float Cv  = dbcb[(size_t)t * NXP + DTR + DS + s];
        h = __expf(dtv * Ads) * h + dtv * uv * Bv;
        float p = h * Cv;                      // reduce across 16 states
        p += __shfl_xor(p, 1, 32);
        p += __shfl_xor(p, 2, 32);
        p += __shfl_xor(p, 4, 32);
        p += __shfl_xor(p, 8, 32);
        if (s == 0) {
            const float zv = zb[(size_t)t * EE];
            ygb[(size_t)t * DI] = (p + uv * Dd) * silu_f(zv);
        }
    }
}

// =====================================================================
// Kernel 6: out_proj  YO[b,l,c] = YG(b,l,:) @ out_proj_w^T
//  weights (64x128, 32KB) async-staged into LDS.
// =====================================================================
__global__ void k_outproj(const float* __restrict__ YG,
                          const float* __restrict__ OW,
                          float* __restrict__ YO) {
    __shared__ float Wl[CC * DI];                 // 32 KB
    const unsigned wbase = lds_off(&Wl[0]);
    for (int i = threadIdx.x * 4; i < CC * DI; i += blockDim.x * 4)
        async_g2l_b128(wbase + i * 4, OW + i);
    wait_async0();
    __syncthreads();

    const int wave = threadIdx.x >> 5;
    const int lane = threadIdx.x & 31;
    const int half = lane >> 4, lm = lane & 15;
    const int tile = blockIdx.x * 8 + wave;
    const int bi = tile >> 10, l0 = (tile & 1023) << 4;

    v8f acc[4];
#pragma unroll
    for (int n = 0; n < 4; ++n) acc[n] = (v8f)0.f;

    const float* Yb = YG + (size_t)bi * LL * DI;
    for (int kk = 0; kk < DI; kk += 4) {
        const int kc = kk + half * 2;
        v2f a;
        a.x = Yb[(size_t)(l0 + lm) * DI + kc];
        a.y = Yb[(size_t)(l0 + lm) * DI + kc + 1];
#pragma unroll
        for (int n = 0; n < 4; ++n) {
            v2f bf;
            bf.x = Wl[(n * 16 + lm) * DI + kc];
            bf.y = Wl[(n * 16 + lm) * DI + kc + 1];
            acc[n] = wmma_k4(a, bf, acc[n]);
        }
    }
#pragma unroll
    for (int n = 0; n < 4; ++n)
#pragma unroll
        for (int r = 0; r < 8; ++r)
            YO[((size_t)bi * LL + l0 + r + half * 8) * CC + n * 16 + lm] = acc[n][r];
}

// =====================================================================
// Kernel 7: GroupNorm statistics. 1 block per (b,g): mean + rstd.
// =====================================================================
__global__ void k_gnstat(const float* __restrict__ YO, float* __restrict__ GN) {
    const int bi = blockIdx.x >> 2, g = blockIdx.x & 3;
    double s = 0.0, s2 = 0.0;
    for (int i = threadIdx.x; i < LL * 16; i += blockDim.x) {
        const int l = i >> 4, c = (i & 15) + g * 16;
        const float v = YO[((size_t)bi * LL + l) * CC + c];
        s += v; s2 += (double)v * v;
    }
    __shared__ double sh[256], sh2[256];
    sh[threadIdx.x] = s; sh2[threadIdx.x] = s2;
    __syncthreads();
    for (int off = 128; off > 0; off >>= 1) {
        if (threadIdx.x < off) {
            sh[threadIdx.x]  += sh[threadIdx.x + off];
            sh2[threadIdx.x] += sh2[threadIdx.x + off];
        }
        __syncthreads();
    }
    if (threadIdx.x == 0) {
        const double n  = (double)LL * 16.0;
        const double mu = sh[0] / n;
        const double var = sh2[0] / n - mu * mu;
        GN[blockIdx.x * 2 + 0] = (float)mu;
        GN[blockIdx.x * 2 + 1] = (float)(1.0 / sqrt(var + (double)GEPS));
    }
}

// =====================================================================
// Kernel 8: normalize + affine + SiLU + residual -> OUT (b,c,h,w)
// =====================================================================
__global__ void k_final(const float* __restrict__ YO,
                        const float* __restrict__ X,
                        const float* __restrict__ GN,
                        const float* __restrict__ gsc,
                        const float* __restrict__ gbi,
                        float* __restrict__ OUT) {
    const int idx = blockIdx.x * blockDim.x + threadIdx.x;
    if (idx >= BB * CC * LL) return;
    const int l  = idx & (LL - 1);
    const int c  = (idx >> 14) & (CC - 1);
    const int bi = idx >> 20;
    const int g  = c >> 4;
    const float mu = GN[(bi * 4 + g) * 2 + 0];
    const float rs = GN[(bi * 4 + g) * 2 + 1];
    const float v  = YO[((size_t)bi * LL + l) * CC + c];
    const float xn = (v - mu) * rs * gsc[c] + gbi[c];
    OUT[idx] = silu_f(xn) + X[idx];
}

// =====================================================================
extern "C" void kernel_launch(void* const* d_in, const int* in_sizes, int n_in,
                              void* d_out, int out_size, void* d_ws, size_t ws_size,
                              hipStream_t stream) {
    const float* x     = (const float*)d_in[0];
    const float* ipw   = (const float*)d_in[1];
    const float* cw    = (const float*)d_in[2];
    const float* cb    = (const float*)d_in[3];
    const float* xpw   = (const float*)d_in[4];
    const float* dtw   = (const float*)d_in[5];
    const float* dtb   = (const float*)d_in[6];
    const float* alog  = (const float*)d_in[7];
    const float* Dp    = (const float*)d_in[8];
    const float* opw   = (const float*)d_in[9];
    const float* gsc   = (const float*)d_in[10];
    const float* gbi   = (const float*)d_in[11];
    float* out = (float*)d_out;
    float* ws  = (float*)d_ws;

    float* XZ  = ws + OFF_XZ;
    float* U   = ws + OFF_U;
    float* DBC = ws + OFF_DBC;
    float* DT  = ws + OFF_DT;
    float* YG  = ws + OFF_YG;
    float* YO  = ws + OFF_YO;
    float* GN  = ws + OFF_GN;

    k_inproj <<<1024,  256, 0, stream>>>(x, ipw, XZ);
    k_conv   <<<16384, 256, 0, stream>>>(XZ, cw, cb, U);
    k_xproj  <<<256,   256, 0, stream>>>(U, xpw, DBC);
    k_dtproj <<<256,   256, 0, stream>>>(DBC, dtw, dtb, DT);
    k_scan   <<<16,    256, 0, stream>>>(DT, U, DBC, XZ, alog, Dp, YG);
    k_outproj<<<256,   256, 0, stream>>>(YG, opw, YO);
    k_gnstat <<<8,     256, 0, stream>>>(YO, GN);
    k_final  <<<8192,  256, 0, stream>>>(YO, x, GN, gsc, gbi, out);
}